// KernelMixingLayer_65481071410654
// MI455X (gfx1250) — compile-verified
//
#include <hip/hip_runtime.h>
#include <hip/hip_bf16.h>

typedef __attribute__((ext_vector_type(2))) float v2f;
typedef __attribute__((ext_vector_type(8))) float v8f;

#define DIM 4096

// ---------------------------------------------------------------------------
// CDNA5 async global->LDS DMA (tracked by ASYNCcnt, no VGPR round trip).
//   global_load_async_to_lds_b128 vdst(lds addr), vaddr(64b global), off
// ---------------------------------------------------------------------------
__device__ __forceinline__ void async_copy_b128(const float* g, const float* l) {
    unsigned int lds  = (unsigned int)(unsigned long long)(uintptr_t)l;
    unsigned long long ga = (unsigned long long)(uintptr_t)g;
    asm volatile("global_load_async_to_lds_b128 %0, %1, off"
                 :
                 : "v"(lds), "v"(ga)
                 : "memory");
}

__device__ __forceinline__ void wait_async() {
    asm volatile("s_wait_asynccnt 0" ::: "memory");
}

// ---------------------------------------------------------------------------
// Kernel 0: per-row squared L2 norms: sq[i] = sum_k x[i,k]^2
// ---------------------------------------------------------------------------
__global__ __launch_bounds__(256) void row_sq_kernel(const float* __restrict__ x,
                                                     float* __restrict__ sq) {
    __shared__ float red[256];
    const int row = blockIdx.x;
    const int tid = threadIdx.x;
    const float* p = x + (size_t)row * DIM;
    float s = 0.0f;
    for (int j = tid; j < DIM; j += 256) {
        float t = p[j];
        s += t * t;
    }
    red[tid] = s;
    __syncthreads();
    for (int off = 128; off > 0; off >>= 1) {
        if (tid < off) red[tid] += red[tid + off];
        __syncthreads();
    }
    if (tid == 0) sq[row] = red[0];
}

// ---------------------------------------------------------------------------
// Templated fp32 WMMA GEMM:  C[M,N] = A[M,K] * B  (+ fused epilogue)
//   BT == true : logical B[k][n] = Bsrc[n*K + k]  (Bsrc row-major [N,K]);
//                B tile staged [n][k] in LDS (same layout as A).
//   BT == false: logical B[k][n] = Bsrc[k*N + n]  (Bsrc row-major [K,N]);
//                B tile staged [k][n] in LDS (no transpose needed).
// EPI: 0 = plain store
//      1 = RBF gram: C' = exp(-|aux1[i,j]| * sqrt(relu(aux0[i]+aux0[j]-2C)))
//      2 = bias:     C' = C + aux0[col]
//
// 256 threads (8 wave32), 128x128 C tile, BK=32, double-buffered LDS with
// async global->LDS DMA prefetch of tile t+1 overlapping WMMA on tile t.
// Each wave: 64x32 sub-tile = 4x2 accum tiles, V_WMMA_F32_16X16X4_F32.
// ---------------------------------------------------------------------------
template <int EPI, bool BT>
__global__ __launch_bounds__(256) void gemm_wmma(const float* __restrict__ A,
                                                 const float* __restrict__ Bsrc,
                                                 float* __restrict__ C,
                                                 const float* __restrict__ aux0,
                                                 const float* __restrict__ aux1) {
    constexpr int BM = 128, BN = 128, BK = 32;
    constexpr int LDA  = BK + 4;                   // 36 floats (144B rows, 16B aligned)
    constexpr int LDB  = BT ? (BK + 4) : (BN + 4); // 36 or 132 (528B rows, 16B aligned)
    constexpr int SBSZ = BT ? (BN * LDB) : (BK * LDB);

    __shared__ float sA[2][BM * LDA];
    __shared__ float sB[2][SBSZ];

    const int tid  = threadIdx.x;
    const int wid  = tid >> 5;
    const int lane = tid & 31;
    const int lh   = lane & 15;   // lane within half
    const int hi   = lane >> 4;   // 0: K pair {0,1}, 1: K pair {2,3}
    const int wm   = wid >> 2;    // 0..1 (row group of 64)
    const int wn   = wid & 3;     // 0..3 (col group of 32)
    const int row0 = blockIdx.y * BM;
    const int col0 = blockIdx.x * BN;

    // ---- async DMA stage of one K-tile into LDS buffer `buf` (issue only) ----
    auto stage = [&](int k0, int buf) {
#pragma unroll
        for (int i = 0; i < 4; ++i) {              // A: 128x32 floats = 1024 x b128
            int f = tid + i * 256;
            int r = f >> 3;
            int c = (f & 7) << 2;
            async_copy_b128(A + (size_t)(row0 + r) * DIM + k0 + c,
                            &sA[buf][r * LDA + c]);
        }
        if constexpr (BT) {
#pragma unroll
            for (int i = 0; i < 4; ++i) {          // B: [n][k] tile, direct copy
                int f = tid + i * 256;
                int r = f >> 3;
                int c = (f & 7) << 2;
                async_copy_b128(Bsrc + (size_t)(col0 + r) * DIM + k0 + c,
                                &sB[buf][r * LDB + c]);
            }
        } else {
#pragma unroll
            for (int i = 0; i < 4; ++i) {          // B: [k][n] tile, direct copy
                int f = tid + i * 256;             // 32 k-rows x 32 n-quads
                int r = f >> 5;
                int c = (f & 31) << 2;
                async_copy_b128(Bsrc + (size_t)(k0 + r) * DIM + col0 + c,
                                &sB[buf][r * LDB + c]);
            }
        }
    };

    v8f acc[4][2];
#pragma unroll
    for (int mi = 0; mi < 4; ++mi)
#pragma unroll
        for (int ni = 0; ni < 2; ++ni) acc[mi][ni] = v8f{};

    // prologue: DMA tile 0
    stage(0, 0);
    wait_async();
    __syncthreads();

    int buf = 0;
    for (int k0 = 0; k0 < DIM; k0 += BK) {
        // prefetch next tile while computing on current one
        if (k0 + BK < DIM) stage(k0 + BK, buf ^ 1);

#pragma unroll
        for (int kk = 0; kk < BK; kk += 4) {
            v2f af[4], bf[2];
#pragma unroll
            for (int mi = 0; mi < 4; ++mi)
                af[mi] = *(const v2f*)(&sA[buf][(wm * 64 + mi * 16 + lh) * LDA + kk + 2 * hi]);
#pragma unroll
            for (int ni = 0; ni < 2; ++ni) {
                if constexpr (BT) {
                    bf[ni] = *(const v2f*)(&sB[buf][(wn * 32 + ni * 16 + lh) * LDB + kk + 2 * hi]);
                } else {
                    const int n = wn * 32 + ni * 16 + lh;
                    v2f b;
                    b.x = sB[buf][(kk + 2 * hi + 0) * LDB + n];
                    b.y = sB[buf][(kk + 2 * hi + 1) * LDB + n];
                    bf[ni] = b;
                }
            }
#pragma unroll
            for (int mi = 0; mi < 4; ++mi)
#pragma unroll
                for (int ni = 0; ni < 2; ++ni)
                    acc[mi][ni] = __builtin_amdgcn_wmma_f32_16x16x4_f32(
                        false, af[mi], false, bf[ni], (short)0, acc[mi][ni],
                        false, false);
        }

        wait_async();      // our prefetch DMAs have landed
        __syncthreads();   // everyone done computing from `buf` + everyone's DMA done
        buf ^= 1;
    }

    // ---- epilogue + store ----
#pragma unroll
    for (int mi = 0; mi < 4; ++mi) {
#pragma unroll
        for (int ni = 0; ni < 2; ++ni) {
            const int gcol = col0 + wn * 32 + ni * 16 + lh;
            float colv = 0.0f;
            if constexpr (EPI == 1) colv = aux0[gcol];   // sq[j]
            if constexpr (EPI == 2) colv = aux0[gcol];   // bias[j]
#pragma unroll
            for (int r = 0; r < 8; ++r) {
                const int grow = row0 + wm * 64 + mi * 16 + hi * 8 + r;
                float v = acc[mi][ni][r];
                if constexpr (EPI == 1) {
                    float d2 = aux0[grow] + colv - 2.0f * v;
                    float d  = d2 > 0.0f ? __builtin_sqrtf(d2) : 0.0f;
                    float w  = __builtin_fabsf(aux1[(size_t)grow * DIM + gcol]);
                    v = __expf(-w * d);
                } else if constexpr (EPI == 2) {
                    v += colv;
                }
                C[(size_t)grow * DIM + gcol] = v;
            }
        }
    }
}

// ---------------------------------------------------------------------------
// Launch:  x -> sq -> gram (fused RBF) -> mixed -> out
// ---------------------------------------------------------------------------
extern "C" void kernel_launch(void* const* d_in, const int* in_sizes, int n_in,
                              void* d_out, int out_size, void* d_ws, size_t ws_size,
                              hipStream_t stream) {
    const float* x     = (const float*)d_in[0];
    const float* W_rbf = (const float*)d_in[1];
    const float* W_out = (const float*)d_in[2];
    const float* b_out = (const float*)d_in[3];
    float* out = (float*)d_out;

    char* ws = (char*)d_ws;
    float* sq    = (float*)ws;                                   // 4096 floats
    float* gram  = (float*)(ws + (64 << 10));                    // 64 MB
    float* mixed = (float*)(ws + (64 << 10) + (size_t)DIM * DIM * sizeof(float));

    dim3 grid(DIM / 128, DIM / 128), block(256);

    row_sq_kernel<<<DIM, 256, 0, stream>>>(x, sq);
    // gram = exp(-|W_rbf| * cdist(x,x)) ; B = x^T  (x is [N,K] row-major -> BT)
    gemm_wmma<1, true><<<grid, block, 0, stream>>>(x, x, gram, sq, W_rbf);
    // mixed = gram @ x ; x is [K,N] row-major -> no transpose
    gemm_wmma<0, false><<<grid, block, 0, stream>>>(gram, x, mixed, nullptr, nullptr);
    // out = mixed @ W_out^T + b ; W_out is [N,K] row-major -> BT
    gemm_wmma<2, true><<<grid, block, 0, stream>>>(mixed, W_out, out, b_out, nullptr);
}